// _MultiTimeAttention_80092550135875
// MI455X (gfx1250) — compile-verified
//
#include <hip/hip_runtime.h>
#include <hip/hip_bf16.h>
#include <math.h>

// Problem constants (from reference)
#define BB   4
#define LQ   128
#define LK   512
#define ET   128   // EMBED_TIME
#define NH   8     // NUM_HEADS
#define EK   16    // ET / NH
#define DD   64    // INPUT_DIM
#define HID  256   // HIDDEN_DIM

typedef __attribute__((ext_vector_type(2))) float v2f;
typedef __attribute__((ext_vector_type(8))) float v8f;

// ---------------------------------------------------------------------------
// Generic batched GEMM on V_WMMA_F32_16X16X4_F32.
//   C[bz] = scale * (A[bz] @ B[bz]) + bias
// One wave (32 lanes) computes one 16x16 tile; blockDim = (32,4) -> 4 n-tiles
// per block. Batch base offsets: off = (bz/div)*sOuter + (bz%div)*sInner,
// which lets a (b,h) batch index address per-head column slices.
// B is addressed as B[k*bsK + n*bsN] so a transposed operand is just a stride
// swap (used for Q·K^T scores).
// ---------------------------------------------------------------------------
template <int KK>
__global__ void gemm_wmma_f32(const float* __restrict__ A, int lda, int divA, long sAo, long sAi,
                              const float* __restrict__ Bm, int bsK, int bsN, int divB, long sBo, long sBi,
                              float* __restrict__ C, int ldc, long sC,
                              const float* __restrict__ bias, float scale)
{
    const int lane = threadIdx.x;            // 0..31, one wave per y-slice
    const int r    = lane & 15;              // row (A/C) or col (B/C) within tile
    const int hi   = lane >> 4;              // selects K pair {2*hi, 2*hi+1}
    const int bz   = blockIdx.z;

    const long tm = (long)blockIdx.x * 16;
    const long tn = ((long)blockIdx.y * 4 + threadIdx.y) * 16;

    const float* __restrict__ Ab = A  + (long)(bz / divA) * sAo + (long)(bz % divA) * sAi;
    const float* __restrict__ Bb = Bm + (long)(bz / divB) * sBo + (long)(bz % divB) * sBi;
    float*       __restrict__ Cb = C  + (long)bz * sC;

    v8f acc = {};
#pragma unroll 8
    for (int k0 = 0; k0 < KK; k0 += 4) {
        const int kk = k0 + 2 * hi;
        v2f a;
        a.x = Ab[(tm + r) * lda + kk];
        a.y = Ab[(tm + r) * lda + kk + 1];
        v2f b;
        b.x = Bb[(long)kk * bsK + (tn + r) * bsN];
        b.y = Bb[(long)(kk + 1) * bsK + (tn + r) * bsN];
        // D = A(16x4) * B(4x16) + C  (f32 WMMA, wave32)
        acc = __builtin_amdgcn_wmma_f32_16x16x4_f32(false, a, false, b,
                                                    (short)0, acc, false, false);
    }

    // C/D layout: VGPR i, lanes 0-15 -> M=i, lanes 16-31 -> M=i+8; N = lane%16
#pragma unroll
    for (int i = 0; i < 8; ++i) {
        const long row = tm + i + 8 * hi;
        const long col = tn + r;
        float v = acc[i] * scale;
        if (bias) v += bias[col];
        Cb[row * ldc + col] = v;
    }
}

// ---------------------------------------------------------------------------
// Per-row (b,h,q): m = max_k s[q,k]; s <- exp(s - m).  One wave per row.
// Masked entries of the reference evaluate to exp(-10000-m) == 0 in f32, so a
// single unmasked row-max reproduces the per-d masked softmax exactly after
// the Z/num GEMMs below.
// ---------------------------------------------------------------------------
__global__ void rowmax_exp_kernel(float* __restrict__ E, int rowlen)
{
    float* p = E + (long)blockIdx.x * rowlen;
    const int lane = threadIdx.x;
    float m = -INFINITY;
    for (int k = lane; k < rowlen; k += 32) m = fmaxf(m, p[k]);
#pragma unroll
    for (int off = 16; off > 0; off >>= 1)
        m = fmaxf(m, __shfl_xor(m, off, 32));
    for (int k = lane; k < rowlen; k += 32)
        p[k] = __expf(p[k] - m);
}

// Mcat[b, k, 0:64]   = mask ? 1 : 0
// Mcat[b, k, 64:128] = mask ? value : 0
__global__ void prep_mask_kernel(const int* __restrict__ mask,
                                 const float* __restrict__ value,
                                 float* __restrict__ Mcat, int total)
{
    int idx = blockIdx.x * blockDim.x + threadIdx.x;
    if (idx >= total) return;
    const int d  = idx & (DD - 1);
    const int bk = idx >> 6;                  // b*LK + k
    const float mf = mask[idx] ? 1.0f : 0.0f;
    float* row = Mcat + (long)bk * (2 * DD);
    row[d]      = mf;
    row[DD + d] = mf * value[idx];
}

// Ccat[bh, q, 0:64] = Z, Ccat[bh, q, 64:128] = num.
// ctx[b, q, h*64 + d] = num / Z   (context, head-major columns)
__global__ void finalize_ctx_kernel(const float* __restrict__ Ccat,
                                    float* __restrict__ ctx, int total)
{
    int idx = blockIdx.x * blockDim.x + threadIdx.x;
    if (idx >= total) return;
    const int d  = idx & (DD - 1);
    const int q  = (idx >> 6) & (LQ - 1);
    const int bh = idx >> 13;                 // b*NH + h
    const int b  = bh >> 3;
    const int h  = bh & 7;
    const float* row = Ccat + ((long)bh * LQ + q) * (2 * DD);
    const float Z   = row[d];
    const float num = row[DD + d];
    ctx[((long)(b * LQ + q)) * (NH * DD) + h * DD + d] = num / Z;
}

extern "C" void kernel_launch(void* const* d_in, const int* in_sizes, int n_in,
                              void* d_out, int out_size, void* d_ws, size_t ws_size,
                              hipStream_t stream) {
    (void)in_sizes; (void)n_in; (void)out_size; (void)ws_size;

    const float* query = (const float*)d_in[0]; // (B, LQ, ET)
    const float* key   = (const float*)d_in[1]; // (B, LK, ET)
    const float* value = (const float*)d_in[2]; // (B, LK, DD)
    const int*   mask  = (const int*)  d_in[3]; // (B, LK, DD)
    const float* Wq    = (const float*)d_in[4]; // (ET, ET)
    const float* bq    = (const float*)d_in[5];
    const float* Wk    = (const float*)d_in[6]; // (ET, ET)
    const float* bk    = (const float*)d_in[7];
    const float* Wo    = (const float*)d_in[8]; // (NH*DD, HID)
    const float* bo    = (const float*)d_in[9];
    float* out = (float*)d_out;                 // (B, LQ, HID)

    // Workspace layout (floats); total ~13.3 MB — lives in the 192 MB L2.
    float* ws   = (float*)d_ws;
    float* Qp   = ws;                                  // B*LQ*ET        =   65536
    float* Kp   = Qp + (long)BB * LQ * ET;             // B*LK*ET        =  262144
    float* E    = Kp + (long)BB * LK * ET;             // B*NH*LQ*LK     = 2097152
    float* Mcat = E  + (long)BB * NH * LQ * LK;        // B*LK*2*DD      =  262144
    float* Ccat = Mcat + (long)BB * LK * 2 * DD;       // B*NH*LQ*2*DD   =  524288
    float* Ctx  = Ccat + (long)BB * NH * LQ * 2 * DD;  // B*LQ*NH*DD     =  262144

    const dim3 blk(32, 4);

    // 1) Qp = query @ Wq + bq      (512 x 128) @ (128 x 128)
    gemm_wmma_f32<ET><<<dim3((BB * LQ) / 16, (ET / 16) / 4, 1), blk, 0, stream>>>(
        query, ET, 1, 0, 0,
        Wq, ET, 1, 1, 0, 0,
        Qp, ET, 0, bq, 1.0f);

    // 2) Kp = key @ Wk + bk        (2048 x 128) @ (128 x 128)
    gemm_wmma_f32<ET><<<dim3((BB * LK) / 16, (ET / 16) / 4, 1), blk, 0, stream>>>(
        key, ET, 1, 0, 0,
        Wk, ET, 1, 1, 0, 0,
        Kp, ET, 0, bk, 1.0f);

    // 3) Scores per (b,h): E = (Qh @ Kh^T) / sqrt(EK).  A base: b*LQ*ET + h*EK;
    //    B is Kp read transposed: element (e, kcol) at Kp[(b*LK+kcol)*ET + h*EK + e].
    gemm_wmma_f32<EK><<<dim3(LQ / 16, (LK / 16) / 4, BB * NH), blk, 0, stream>>>(
        Qp, ET, NH, (long)LQ * ET, EK,
        Kp, 1, ET, NH, (long)LK * ET, EK,
        E, LK, (long)LQ * LK, nullptr, 0.25f);

    // 4) Row-softmax numerators: E <- exp(E - rowmax)   (one wave per row)
    rowmax_exp_kernel<<<dim3(BB * NH * LQ), dim3(32), 0, stream>>>(E, LK);

    // 5) Mcat = [maskf | maskf*value]   (B, LK, 128)
    {
        const int total = BB * LK * DD;
        prep_mask_kernel<<<dim3((total + 255) / 256), dim3(256), 0, stream>>>(
            mask, value, Mcat, total);
    }

    // 6) [Z | num] = E @ Mcat per (b,h):  (128 x 512) @ (512 x 128)
    gemm_wmma_f32<LK><<<dim3(LQ / 16, (2 * DD / 16) / 4, BB * NH), blk, 0, stream>>>(
        E, LK, 1, (long)LQ * LK, 0,
        Mcat, 2 * DD, 1, NH, (long)LK * 2 * DD, 0,
        Ccat, 2 * DD, (long)LQ * 2 * DD, nullptr, 1.0f);

    // 7) ctx[b,q,h*64+d] = num / Z
    {
        const int total = BB * NH * LQ * DD;
        finalize_ctx_kernel<<<dim3((total + 255) / 256), dim3(256), 0, stream>>>(
            Ccat, Ctx, total);
    }

    // 8) out = ctx @ Wo + bo       (512 x 512) @ (512 x 256)
    gemm_wmma_f32<NH * DD><<<dim3((BB * LQ) / 16, (HID / 16) / 4, 1), blk, 0, stream>>>(
        Ctx, NH * DD, 1, 0, 0,
        Wo, HID, 1, 1, 0, 0,
        out, HID, 0, bo, 1.0f);
}